// Net_43224550868235
// MI455X (gfx1250) — compile-verified
//
#include <hip/hip_runtime.h>

typedef __attribute__((ext_vector_type(2))) float v2f;
typedef __attribute__((ext_vector_type(8))) float v8f;

#define ALPHA 0.9f
#define BETA  0.85f
#define THR   1.0f
#define NSTEP 100
#define NB    8192
#define NIN   784
#define NH    1000
#define NOUT  10

// ---------------------------------------------------------------------------
// Kernel A: cur1 = x @ W1^T + b1   (f32 WMMA 16x16x4)
// Block = 128 threads (4 waves); each wave owns one 16x16 output tile.
// Block covers 16 (M) x 64 (N). Grid = (8192/16, ceil(1000/64)) = (512, 16).
// x and W1 are L2-resident (25.7MB + 3.1MB << 192MB L2), so no LDS staging.
// ---------------------------------------------------------------------------
__global__ __launch_bounds__(128) void gemm1_wmma(
    const float* __restrict__ x, const float* __restrict__ W1,
    const float* __restrict__ b1, float* __restrict__ cur1)
{
  const int lane  = threadIdx.x & 31;
  const int wave  = threadIdx.x >> 5;
  const int mBase = blockIdx.x * 16;
  const int nBase = blockIdx.y * 64 + wave * 16;

  const int mr   = lane & 15;          // M index (A) / N index (B) in tile
  const int koff = (lane >> 4) * 2;    // K offset within a K=4 chunk

  // clamp B row so out-of-range N tiles still load in-bounds (store is guarded)
  int nr = nBase + mr; if (nr > NH - 1) nr = NH - 1;

  const float* xrow = x  + (size_t)(mBase + mr) * NIN + koff;
  const float* wrow = W1 + (size_t)nr * NIN + koff;

  v8f acc = {};
  for (int k = 0; k < NIN; k += 4) {   // 196 chunks
    v2f a = *(const v2f*)(xrow + k);
    v2f b = *(const v2f*)(wrow + k);
    acc = __builtin_amdgcn_wmma_f32_16x16x4_f32(
        false, a, false, b, (short)0, acc, false, false);
  }

  // C/D layout: VGPR r, lanes 0-15 -> M=r, N=lane; lanes 16-31 -> M=8+r
  const int n = nBase + mr;
  if (n < NH) {
    const float bias = b1[n];
    const int m = mBase + (lane >> 4) * 8;
#pragma unroll
    for (int r = 0; r < 8; ++r)
      cur1[(size_t)(m + r) * NH + n] = acc[r] + bias;
  }
}

// ---------------------------------------------------------------------------
// Kernel B: fused 100-step synaptic-LIF scan.
// One block (256 threads, 8 waves) per 16 batch rows. syn1/mem1/cur1 live in
// registers (63 elements per thread). spk1 + zero-padded W2 live in LDS.
// W2 is staged with async global->LDS B128 copies (ASYNCcnt path).
// Per step: LIF1 -> barrier -> WMMA (K split over 8 waves) -> barrier ->
// cross-wave reduce + LIF2 + output store -> barrier.
// Dynamic LDS: 16000 (spk) + 16000 (W2 pad to 16 rows) + 2048 (partials) f32.
// ---------------------------------------------------------------------------
#if __has_builtin(__builtin_amdgcn_global_load_async_to_lds_b128) && \
    __has_builtin(__builtin_amdgcn_s_wait_asynccnt)
#define USE_ASYNC_LDS 1
typedef int v4i_vs __attribute__((vector_size(16)));
typedef __attribute__((address_space(1))) v4i_vs* async_gptr;
typedef __attribute__((address_space(3))) v4i_vs* async_lptr;
#else
#define USE_ASYNC_LDS 0
#endif

__global__ __launch_bounds__(256) void snn_scan(
    const float* __restrict__ cur1, const float* __restrict__ W2,
    const float* __restrict__ b2, float* __restrict__ out)
{
  extern __shared__ float smem[];
  float* sSpk  = smem;           // [16][1000]
  float* sW2   = smem + 16000;   // [16][1000], rows >= 10 are zero
  float* sPart = smem + 32000;   // [8][256] partial C fragments

  const int t     = threadIdx.x;
  const int lane  = t & 31;
  const int wave  = t >> 5;
  const int bBase = blockIdx.x * 16;

  // ---- stage W2 rows 0..9 (40000 B) into LDS ----
#if USE_ASYNC_LDS
  // async memory->LDS DMA path (tracked by ASYNCcnt, no VGPR round-trip)
  for (int c = t; c < (NOUT * NH) / 4; c += 256) {       // 2500 x 16B chunks
    __builtin_amdgcn_global_load_async_to_lds_b128(
        (async_gptr)(W2 + 4 * c), (async_lptr)(sW2 + 4 * c), 0, 0);
  }
#else
  for (int i = t; i < NOUT * NH; i += 256) sW2[i] = W2[i];
#endif
  // zero-pad rows 10..15 so the WMMA B fragment needs no guard
  for (int i = NOUT * NH + t; i < 16000; i += 256) sW2[i] = 0.0f;

  // per-thread LIF1 state: flat element e = t + 256*j over [16 x 1000]
  float syn1[63], mem1[63], cur[63];
#pragma unroll
  for (int j = 0; j < 63; ++j) {
    const int e = t + 256 * j;
    const bool ok = (j < 62) || (t < 128);     // e < 16000
    syn1[j] = 0.0f; mem1[j] = 0.0f;
    cur[j] = ok ? cur1[(size_t)(bBase + e / NH) * NH + (e % NH)] : 0.0f;
  }

  // layer-2 state ownership: thread t owns C slot (r = t>>5, l = t&31)
  const int n2   = lane & 15;
  const int m2   = (t >> 5) + ((lane >> 4) << 3);
  const bool own2 = (n2 < NOUT);
  float syn2 = 0.0f, mem2 = 0.0f;
  const float bias2 = own2 ? b2[n2] : 0.0f;

  // WMMA operand base offsets in LDS (same pattern for A and B fragments)
  const int fOff = (lane & 15) * NH + ((lane >> 4) * 2);

#if USE_ASYNC_LDS
  __builtin_amdgcn_s_wait_asynccnt(0);   // W2 tiles resident in LDS
#endif
  __syncthreads();

  for (int step = 0; step < NSTEP; ++step) {
    // ---- LIF layer 1 + spike generation into LDS ----
#pragma unroll
    for (int j = 0; j < 63; ++j) {
      if ((j < 62) || (t < 128)) {
        const float r1 = (mem1[j] > THR) ? 1.0f : 0.0f;   // detached reset
        syn1[j] = ALPHA * syn1[j] + cur[j];
        const float m = BETA * mem1[j] + syn1[j] - r1 * THR;
        mem1[j] = m;
        sSpk[t + 256 * j] = ((m - THR) > 0.0f) ? 1.0f : 0.0f;
      }
    }
    __syncthreads();

    // ---- cur2 = spk1 @ W2^T via f32 WMMA, K=1000 split over 8 waves ----
    v8f acc = {};
    for (int c = wave; c < 250; c += 8) {
      v2f a = *(const v2f*)(sSpk + fOff + 4 * c);
      v2f b = *(const v2f*)(sW2  + fOff + 4 * c);
      acc = __builtin_amdgcn_wmma_f32_16x16x4_f32(
          false, a, false, b, (short)0, acc, false, false);
    }
#pragma unroll
    for (int r = 0; r < 8; ++r)
      sPart[wave * 256 + r * 32 + lane] = acc[r];
    __syncthreads();

    // ---- cross-wave reduce + LIF layer 2 + record outputs ----
    if (own2) {
      float s = 0.0f;
#pragma unroll
      for (int w = 0; w < 8; ++w) s += sPart[w * 256 + t];
      const float cur2 = s + bias2;
      const float r2 = (mem2 > THR) ? 1.0f : 0.0f;
      syn2 = ALPHA * syn2 + cur2;
      mem2 = BETA * mem2 + syn2 - r2 * THR;
      const float spk2 = ((mem2 - THR) > 0.0f) ? 1.0f : 0.0f;
      const size_t gi = (size_t)step * (NB * NOUT) + (size_t)(bBase + m2) * NOUT + n2;
      out[gi] = spk2;                                   // spk_rec
      out[(size_t)NSTEP * NB * NOUT + gi] = mem2;       // mem_rec
    }
    __syncthreads();
  }
}

// ---------------------------------------------------------------------------
extern "C" void kernel_launch(void* const* d_in, const int* in_sizes, int n_in,
                              void* d_out, int out_size, void* d_ws, size_t ws_size,
                              hipStream_t stream) {
  const float* x  = (const float*)d_in[0];
  const float* W1 = (const float*)d_in[1];
  const float* b1 = (const float*)d_in[2];
  const float* W2 = (const float*)d_in[3];
  const float* b2 = (const float*)d_in[4];
  float* out  = (float*)d_out;
  float* cur1 = (float*)d_ws;   // 8192*1000*4 = 32.8 MB scratch

  dim3 gridA(NB / 16, (NH + 63) / 64);   // (512, 16)
  gemm1_wmma<<<gridA, 128, 0, stream>>>(x, W1, b1, cur1);

  const size_t ldsB = (size_t)(16000 + 16000 + 8 * 256) * sizeof(float); // ~133 KB
  snn_scan<<<dim3(NB / 16), 256, ldsB, stream>>>(cur1, W2, b2, out);
}